// RotateNMS_31791347925389
// MI455X (gfx1250) — compile-verified
//
#include <hip/hip_runtime.h>
#include <hip/hip_bf16.h>

#define N_BOXES 2048
#define NMS_THR 0.7f
#define MASK_WORDS_PER_ROW 64   // 2048 bits
#define MASK_TOTAL_WORDS (N_BOXES * MASK_WORDS_PER_ROW)

typedef __attribute__((ext_vector_type(2))) float v2f;
typedef __attribute__((ext_vector_type(8))) float v8f;

// ---------------- workspace layout (bytes) ----------------
// [0, 524288)          : suppression bitmask, uint32 mask[2048][64]
// [524288, 557056)     : prefilter float4 {x, y, r, s} per box
// [557056, 622592)     : corners float[2048][8]  (x0,y0,x1,y1,x2,y2,x3,y3)
// [622592, 630784)     : area float[2048]
#define WS_MASK_OFF    0
#define WS_PREF_OFF    524288
#define WS_CORN_OFF    557056
#define WS_AREA_OFF    622592

// ---------------------------------------------------------------------------
// Kernel: zero the suppression mask
// ---------------------------------------------------------------------------
__global__ void nms_zero_mask(unsigned* __restrict__ mask, int nwords) {
    int i = blockIdx.x * blockDim.x + threadIdx.x;
    if (i < nwords) mask[i] = 0u;
}

// ---------------------------------------------------------------------------
// Kernel: per-box precompute (corners, area, prefilter vector)
// Box layout: [xc, yc, w, h, theta]
// Reference corner order: (+,+), (-,+), (-,-), (+,-) in local frame.
// ---------------------------------------------------------------------------
__global__ void nms_prep(const float* __restrict__ boxes,
                         float4* __restrict__ pref,
                         float* __restrict__ corners,
                         float* __restrict__ area) {
    int i = blockIdx.x * blockDim.x + threadIdx.x;
    if (i >= N_BOXES) return;
    float xc = boxes[i * 5 + 0];
    float yc = boxes[i * 5 + 1];
    float w  = boxes[i * 5 + 2];
    float h  = boxes[i * 5 + 3];
    float t  = boxes[i * 5 + 4];
    float c = cosf(t), s = sinf(t);
    float hw = 0.5f * w, hh = 0.5f * h;
    const float dx[4] = { hw, -hw, -hw,  hw };
    const float dy[4] = { hh,  hh, -hh, -hh };
#pragma unroll
    for (int k = 0; k < 4; ++k) {
        corners[i * 8 + 2 * k + 0] = xc + c * dx[k] - s * dy[k];
        corners[i * 8 + 2 * k + 1] = yc + s * dx[k] + c * dy[k];
    }
    // circumradius + 0.5px safety margin (covers f32 cancellation in s)
    float r = 0.5f * sqrtf(w * w + h * h) + 0.5f;
    pref[i] = make_float4(xc, yc, r, xc * xc + yc * yc - r * r);
    area[i] = w * h;
}

// ---------------------------------------------------------------------------
// Exact rotated IoU via Sutherland-Hodgman clipping (matches reference math).
// c1 = subject quad (box i), c2 = clip quad (box j).
// ---------------------------------------------------------------------------
__device__ float pair_iou(const float* __restrict__ c1,
                          const float* __restrict__ c2,
                          float a1, float a2) {
    float px[8], py[8], qx[8], qy[8];
#pragma unroll
    for (int k = 0; k < 4; ++k) { px[k] = c1[2 * k]; py[k] = c1[2 * k + 1]; }
    int n = 4;
#pragma unroll 1
    for (int e = 0; e < 4; ++e) {
        float ex = c2[2 * e], ey = c2[2 * e + 1];
        int e2 = (e + 1) & 3;
        float dx = c2[2 * e2] - ex;
        float dy = c2[2 * e2 + 1] - ey;
        int m = 0;
        float ppx = px[n - 1], ppy = py[n - 1];
        float sp = dx * (ppy - ey) - dy * (ppx - ex);
#pragma unroll 1
        for (int k = 0; k < n; ++k) {
            float cx = px[k], cy = py[k];
            float sc = dx * (cy - ey) - dy * (cx - ex);
            bool cur_in = (sc >= 0.0f), prev_in = (sp >= 0.0f);
            if (cur_in != prev_in) {
                float tt = sp / (sp - sc);
                qx[m] = ppx + tt * (cx - ppx);
                qy[m] = ppy + tt * (cy - ppy);
                ++m;
            }
            if (cur_in) { qx[m] = cx; qy[m] = cy; ++m; }
            ppx = cx; ppy = cy; sp = sc;
        }
        n = m;
        if (n == 0) break;
#pragma unroll 1
        for (int k = 0; k < n; ++k) { px[k] = qx[k]; py[k] = qy[k]; }
    }
    float inter = 0.0f;
    if (n >= 3) {
        float sx = px[n - 1], sy = py[n - 1];
#pragma unroll 1
        for (int k = 0; k < n; ++k) {
            inter += sx * py[k] - px[k] * sy;
            sx = px[k]; sy = py[k];
        }
        inter = 0.5f * fabsf(inter);
    }
    float uni = a1 + a2 - inter;
    return inter / fmaxf(uni, 1e-8f);
}

// ---------------------------------------------------------------------------
// Kernel: one wave per 16x16 tile of the pair matrix (lower triangle only).
// WMMA computes  D[i][j] = dist(ci,cj)^2 - (ri+rj)^2  for the whole tile:
//   A row i = (x, y, r, s)_i ;  B col j = (-2x, -2y, -2r, 1)_j ;  C = s_j.
// Survivors (D<=0, j<i) run the exact polygon IoU; hits set bits in mask[i].
// ---------------------------------------------------------------------------
__global__ __launch_bounds__(32) void nms_tile(const float4* __restrict__ pref,
                                               const float* __restrict__ corners,
                                               const float* __restrict__ area,
                                               unsigned* __restrict__ mask) {
    const int ti = blockIdx.y;   // row tile
    const int tj = blockIdx.x;   // col tile
    if (tj > ti) return;         // need j < i only (uniform whole-wave exit)

    const int lane = threadIdx.x;        // 0..31, wave32
    const int i0 = ti * 16, j0 = tj * 16;
    const bool hi = (lane >= 16);
    const int  ln = lane & 15;

    float4 pa = pref[i0 + ln];
    float4 pb = pref[j0 + ln];

    // A fragment (16x4 f32): lanes 0-15 hold K=0,1 ; lanes 16-31 hold K=2,3
    v2f A;
    A[0] = hi ? pa.z : pa.x;
    A[1] = hi ? pa.w : pa.y;
    // B fragment (4x16 f32): column = lane%16 ; lanes 0-15 rows K=0,1 ; 16-31 rows K=2,3
    v2f B;
    B[0] = hi ? (-2.0f * pb.z) : (-2.0f * pb.x);
    B[1] = hi ? 1.0f           : (-2.0f * pb.y);
    // C fragment: column = lane%16 is constant per lane -> bias with s_j
    v8f C;
#pragma unroll
    for (int k = 0; k < 8; ++k) C[k] = pb.w;

    v8f D = __builtin_amdgcn_wmma_f32_16x16x4_f32(
        /*neg_a=*/false, A, /*neg_b=*/false, B,
        /*c_mod=*/(short)0, C, /*reuse_a=*/false, /*reuse_b=*/false);

    // C/D layout: lane l, reg k -> row i0 + k + 8*(l/16), col j0 + (l%16)
    const int j = j0 + ln;
    const int ibase = i0 + (hi ? 8 : 0);
#pragma unroll 1
    for (int k = 0; k < 8; ++k) {
        int i = ibase + k;
        if (j < i && D[k] <= 0.0f) {
            float iou = pair_iou(corners + i * 8, corners + j * 8, area[i], area[j]);
            if (iou > NMS_THR) {
                atomicOr(&mask[i * MASK_WORDS_PER_ROW + (j >> 5)], 1u << (j & 31));
            }
        }
    }
}

// ---------------------------------------------------------------------------
// Kernel: sequential greedy NMS scan (single wave) + compaction.
// keep bitset lives in LDS; next mask row is prefetched to hide latency.
// Output: sorted kept indices, then -1 padding (int32, 2048 entries).
// ---------------------------------------------------------------------------
__global__ __launch_bounds__(32) void nms_scan(const unsigned* __restrict__ mask,
                                               int* __restrict__ out) {
    __shared__ unsigned keepw[64];
    __shared__ int prefix[64];
    __shared__ int totalS;
    const int lane = threadIdx.x;

    keepw[lane] = 0u;
    keepw[lane + 32] = 0u;
    __syncthreads();

    unsigned m0 = mask[lane];
    unsigned m1 = mask[32 + lane];
#pragma unroll 1
    for (int i = 0; i < N_BOXES; ++i) {
        unsigned n0 = 0u, n1 = 0u;
        if (i + 1 < N_BOXES) {                      // prefetch next row
            n0 = mask[(i + 1) * MASK_WORDS_PER_ROW + lane];
            n1 = mask[(i + 1) * MASK_WORDS_PER_ROW + 32 + lane];
        }
        unsigned s = (m0 & keepw[lane]) | (m1 & keepw[lane + 32]);
        bool suppressed = __any(s != 0u);
        if (!suppressed && lane == 0) keepw[i >> 5] |= (1u << (i & 31));
        __syncthreads();
        m0 = n0; m1 = n1;
    }

    if (lane == 0) {
        int p = 0;
#pragma unroll 1
        for (int w = 0; w < 64; ++w) { prefix[w] = p; p += __popc(keepw[w]); }
        totalS = p;
    }
    __syncthreads();

#pragma unroll 1
    for (int w = lane; w < 64; w += 32) {
        unsigned bits = keepw[w];
        int p = prefix[w];
        while (bits) {
            int b = __ffs(bits) - 1;
            out[p++] = w * 32 + b;
            bits &= bits - 1;
        }
    }
    int total = totalS;
    for (int p = total + lane; p < N_BOXES; p += 32) out[p] = -1;
}

// ---------------------------------------------------------------------------
extern "C" void kernel_launch(void* const* d_in, const int* in_sizes, int n_in,
                              void* d_out, int out_size, void* d_ws, size_t ws_size,
                              hipStream_t stream) {
    const float* boxes = (const float*)d_in[0];
    int* out = (int*)d_out;    // reference output dtype is int32 (keep indices / -1)
    char* ws = (char*)d_ws;

    unsigned* mask   = (unsigned*)(ws + WS_MASK_OFF);
    float4*   pref   = (float4*)(ws + WS_PREF_OFF);
    float*    corn   = (float*)(ws + WS_CORN_OFF);
    float*    area   = (float*)(ws + WS_AREA_OFF);

    nms_zero_mask<<<(MASK_TOTAL_WORDS + 255) / 256, 256, 0, stream>>>(mask, MASK_TOTAL_WORDS);
    nms_prep<<<(N_BOXES + 255) / 256, 256, 0, stream>>>(boxes, pref, corn, area);

    dim3 grid(N_BOXES / 16, N_BOXES / 16);   // 128 x 128 tiles, upper triangle exits
    nms_tile<<<grid, 32, 0, stream>>>(pref, corn, area, mask);

    nms_scan<<<1, 32, 0, stream>>>(mask, out);
}